// HeteroGNN_3659312136510
// MI455X (gfx1250) — compile-verified
//
#include <hip/hip_runtime.h>
#include <hip/hip_bf16.h>

typedef __attribute__((ext_vector_type(2))) float v2f;
typedef __attribute__((ext_vector_type(8))) float v8f;

#define FD 64           // feature dim (in & hidden)
#define RTY 3           // relations
#define WP_STRIDE 160   // padded float stride per kp-row in LDS (bank-conflict free)

// ---------------------------------------------------------------------------
// Utility: zero a float region
// ---------------------------------------------------------------------------
__global__ __launch_bounds__(256) void zero_f32(float* __restrict__ p, int n) {
    int i = blockIdx.x * 256 + threadIdx.x;
    if (i < n) p[i] = 0.0f;
}

// ---------------------------------------------------------------------------
// Degree counting: deg_out[r][src], deg_in[r][dst] as uint counts.
// ---------------------------------------------------------------------------
__global__ __launch_bounds__(256) void deg_count(
    const int* __restrict__ src, const int* __restrict__ dst,
    unsigned* __restrict__ degout, unsigned* __restrict__ degin,
    int n_edges_per_rel, int total, int n_nodes) {
    int i = blockIdx.x * 256 + threadIdx.x;
    if (i >= total) return;
    int r = i / n_edges_per_rel;
    atomicAdd(&degout[r * n_nodes + src[i]], 1u);
    atomicAdd(&degin [r * n_nodes + dst[i]], 1u);
}

// In-place: uint count -> rsqrt(max(count,1))
__global__ __launch_bounds__(256) void make_scale(float* __restrict__ buf, int n) {
    int i = blockIdx.x * 256 + threadIdx.x;
    if (i >= n) return;
    unsigned c = ((const unsigned*)buf)[i];
    float d = (float)(c < 1u ? 1u : c);
    buf[i] = rsqrtf(d);
}

// ---------------------------------------------------------------------------
// H = (X * s_out[row]) @ W   via V_WMMA_F32_16X16X4_F32
// X: [n x 64], W: [64 x 64] row-major (k-major), H: [n x 64]
// Block = 128 threads = 4 waves; each wave owns a 16x64 output tile.
// W staged in LDS with k-pairs interleaved: Wl[(k>>1)*160 + col*2 + (k&1)],
// so each B fragment is ONE aligned ds_load_b64 (no repacking movs), and the
// 160-float kp-row stride puts the two half-waves in disjoint bank sets.
// Requires n % 16 == 0 (100000 = 6250 * 16).
// ---------------------------------------------------------------------------
__global__ __launch_bounds__(128) void gemm_scaled_wmma(
    const float* __restrict__ X, const float* __restrict__ W,
    const float* __restrict__ sout, float* __restrict__ H, int n_nodes) {
    __shared__ float Wl[(FD / 2) * WP_STRIDE / 1];  // 32 * 160 floats = 20 KB

    // cooperative load + pair-interleave of 64x64 fp32 weights
    for (int idx = threadIdx.x; idx < FD * FD; idx += 128) {
        int k = idx >> 6, col = idx & 63;
        Wl[(k >> 1) * WP_STRIDE + col * 2 + (k & 1)] = W[idx];
    }
    __syncthreads();

    const int wave = threadIdx.x >> 5;
    const int lane = threadIdx.x & 31;
    const int hi   = lane >> 4;     // half-wave index (0/1)
    const int t    = lane & 15;     // row (A) / col (B,C) index within tile

    const int row0 = blockIdx.x * 64 + wave * 16;
    if (row0 >= n_nodes) return;    // whole-wave uniform -> EXEC stays all-1s

    const int gm = row0 + t;                       // this lane's A row
    const float sc = sout[gm];                     // rsqrt(out-degree) row scale
    const float* __restrict__ xrow = X + (size_t)gm * FD;
    const v2f* __restrict__ Wp = (const v2f*)Wl;   // pair stride per kp-row = 80

    v8f acc0 = {}, acc1 = {}, acc2 = {}, acc3 = {};

#pragma unroll
    for (int k0 = 0; k0 < FD; k0 += 4) {
        const int kb = k0 + 2 * hi;
        v2f a;
        a.x = xrow[kb]     * sc;
        a.y = xrow[kb + 1] * sc;
        const v2f* wrow = Wp + ((k0 >> 1) + hi) * (WP_STRIDE / 2) + t;
        v2f b0 = wrow[0];
        v2f b1 = wrow[16];
        v2f b2 = wrow[32];
        v2f b3 = wrow[48];
        acc0 = __builtin_amdgcn_wmma_f32_16x16x4_f32(false, a, false, b0, (short)0, acc0, false, false);
        acc1 = __builtin_amdgcn_wmma_f32_16x16x4_f32(false, a, false, b1, (short)0, acc1, false, false);
        acc2 = __builtin_amdgcn_wmma_f32_16x16x4_f32(false, a, false, b2, (short)0, acc2, false, false);
        acc3 = __builtin_amdgcn_wmma_f32_16x16x4_f32(false, a, false, b3, (short)0, acc3, false, false);
    }

    // C/D layout: VGPR i -> M = i + 8*hi, N = t
    float* out0 = H + (size_t)(row0 + 8 * hi) * FD + t;
#pragma unroll
    for (int i = 0; i < 8; ++i) {
        float* orow = out0 + (size_t)i * FD;
        orow[0]  = acc0[i];
        orow[16] = acc1[i];
        orow[32] = acc2[i];
        orow[48] = acc3[i];
    }
}

// ---------------------------------------------------------------------------
// Edge scatter: Y[dst] += H[src] * s_in[dst].  One wave per EDGES_PER_WAVE
// edges; each edge's 64 floats handled as float2 per lane.  Index loads and
// H-row gathers for all edges issued before any atomics (memory-level
// parallelism); dest-side rsqrt(deg_in) scale folded in so one accumulator
// serves all relations.
// ---------------------------------------------------------------------------
#define EDGES_PER_WAVE 4
__global__ __launch_bounds__(256) void scatter_edges(
    const float* __restrict__ H, const int* __restrict__ src,
    const int* __restrict__ dst, const float* __restrict__ sin_,
    float* __restrict__ Y, int n_edges) {
    const int lane = threadIdx.x & 31;
    const int wid = blockIdx.x * (blockDim.x >> 5) + (threadIdx.x >> 5);
    const int e0 = wid * EDGES_PER_WAVE;
    if (e0 >= n_edges) return;

    if (e0 + EDGES_PER_WAVE <= n_edges) {
        int s[EDGES_PER_WAVE], d[EDGES_PER_WAVE];
        float f[EDGES_PER_WAVE];
        float2 v[EDGES_PER_WAVE];
#pragma unroll
        for (int i = 0; i < EDGES_PER_WAVE; ++i) {
            s[i] = src[e0 + i];
            d[i] = dst[e0 + i];
        }
#pragma unroll
        for (int i = 0; i < EDGES_PER_WAVE; ++i) {
            f[i] = sin_[d[i]];
            v[i] = ((const float2*)(H + (size_t)s[i] * FD))[lane];
        }
#pragma unroll
        for (int i = 0; i < EDGES_PER_WAVE; ++i) {
            float* yp = Y + (size_t)d[i] * FD + 2 * lane;
            unsafeAtomicAdd(yp,     v[i].x * f[i]);
            unsafeAtomicAdd(yp + 1, v[i].y * f[i]);
        }
    } else {
        for (int e = e0; e < n_edges; ++e) {
            const int s = src[e];
            const int d = dst[e];
            const float f = sin_[d];
            const float2 v = ((const float2*)(H + (size_t)s * FD))[lane];
            float* yp = Y + (size_t)d * FD + 2 * lane;
            unsafeAtomicAdd(yp,     v.x * f);
            unsafeAtomicAdd(yp + 1, v.y * f);
        }
    }
}

// X = relu(Y + sum_r b[r])
__global__ __launch_bounds__(256) void bias_relu(
    const float* __restrict__ Y, const float* __restrict__ b /*[3][64]*/,
    float* __restrict__ X, int n) {
    int i = blockIdx.x * 256 + threadIdx.x;
    if (i >= n) return;
    int j = i & (FD - 1);
    float bb = b[j] + b[FD + j] + b[2 * FD + j];
    float v = Y[i] + bb;
    X[i] = v > 0.0f ? v : 0.0f;
}

// ---------------------------------------------------------------------------
// Final layer (dout = 2): tiny GEMM per node + scatter + bias
// ---------------------------------------------------------------------------
__global__ __launch_bounds__(256) void gemm2(
    const float* __restrict__ X, const float* __restrict__ W /*64x2*/,
    const float* __restrict__ sout, float* __restrict__ h4, int n_nodes) {
    int node = blockIdx.x * 256 + threadIdx.x;
    if (node >= n_nodes) return;
    float sc = sout[node];
    const float* xr = X + (size_t)node * FD;
    float a0 = 0.f, a1 = 0.f;
#pragma unroll
    for (int k = 0; k < FD; ++k) {
        float xv = xr[k] * sc;
        a0 += xv * W[2 * k];
        a1 += xv * W[2 * k + 1];
    }
    h4[2 * node]     = a0;
    h4[2 * node + 1] = a1;
}

__global__ __launch_bounds__(256) void scatter2(
    const float* __restrict__ h4, const int* __restrict__ src,
    const int* __restrict__ dst, const float* __restrict__ sin_,
    float* __restrict__ out, int n_edges) {
    int e = blockIdx.x * 256 + threadIdx.x;
    if (e >= n_edges) return;
    int s = src[e], d = dst[e];
    float f = sin_[d];
    unsafeAtomicAdd(&out[2 * d],     h4[2 * s] * f);
    unsafeAtomicAdd(&out[2 * d + 1], h4[2 * s + 1] * f);
}

__global__ __launch_bounds__(256) void bias2(
    float* __restrict__ out, const float* __restrict__ b4 /*[3][2]*/, int n) {
    int i = blockIdx.x * 256 + threadIdx.x;
    if (i >= n) return;
    int c = i & 1;
    out[i] += b4[c] + b4[2 + c] + b4[4 + c];
}

// ---------------------------------------------------------------------------
// Host-side orchestration
// ---------------------------------------------------------------------------
extern "C" void kernel_launch(void* const* d_in, const int* in_sizes, int n_in,
                              void* d_out, int out_size, void* d_ws, size_t ws_size,
                              hipStream_t stream) {
    const float* x   = (const float*)d_in[0];
    const int*   src = (const int*)  d_in[1];
    const int*   dst = (const int*)  d_in[2];
    const float* Wl[3] = { (const float*)d_in[3], (const float*)d_in[5], (const float*)d_in[7] };
    const float* bl[3] = { (const float*)d_in[4], (const float*)d_in[6], (const float*)d_in[8] };
    const float* W4 = (const float*)d_in[9];
    const float* b4 = (const float*)d_in[10];

    const int N = in_sizes[0] / FD;        // 100000
    const int E = in_sizes[1] / RTY;       // 800000

    // workspace layout (floats)
    float* ws    = (float*)d_ws;
    float* s_out = ws;                      // [3*N]   (also uint deg scratch)
    float* s_in  = ws + (size_t)RTY * N;    // [3*N]
    float* Xb    = ws + (size_t)2 * RTY * N;          // [N*64]
    float* Hb    = Xb + (size_t)N * FD;               // [N*64]
    float* Yb    = Hb + (size_t)N * FD;               // [N*64]
    float* h4    = Yb + (size_t)N * FD;               // [N*2]
    float* outp  = (float*)d_out;

    const int scaleN = 2 * RTY * N;                   // 600000
    // 1) degrees -> rsqrt scales (recomputed every call: deterministic)
    zero_f32<<<(scaleN + 255) / 256, 256, 0, stream>>>(s_out, scaleN);
    deg_count<<<(RTY * E + 255) / 256, 256, 0, stream>>>(
        src, dst, (unsigned*)s_out, (unsigned*)s_in, E, RTY * E, N);
    make_scale<<<(scaleN + 255) / 256, 256, 0, stream>>>(s_out, scaleN);

    const int gemmBlocks    = (N + 63) / 64;
    const int scatterBlocks = (E + 8 * EDGES_PER_WAVE - 1) / (8 * EDGES_PER_WAVE);

    // 2) three hidden hetero-conv layers (64 -> 64, ReLU)
    const float* cur = x;
    for (int l = 0; l < 3; ++l) {
        zero_f32<<<((size_t)N * FD + 255) / 256, 256, 0, stream>>>(Yb, N * FD);
        for (int r = 0; r < RTY; ++r) {
            gemm_scaled_wmma<<<gemmBlocks, 128, 0, stream>>>(
                cur, Wl[l] + (size_t)r * FD * FD, s_out + (size_t)r * N, Hb, N);
            scatter_edges<<<scatterBlocks, 256, 0, stream>>>(
                Hb, src + (size_t)r * E, dst + (size_t)r * E,
                s_in + (size_t)r * N, Yb, E);
        }
        bias_relu<<<((size_t)N * FD + 255) / 256, 256, 0, stream>>>(Yb, bl[l], Xb, N * FD);
        cur = Xb;
    }

    // 3) final hetero-conv layer (64 -> 2 logits, no ReLU)
    zero_f32<<<(2 * N + 255) / 256, 256, 0, stream>>>(outp, 2 * N);
    for (int r = 0; r < RTY; ++r) {
        gemm2<<<(N + 255) / 256, 256, 0, stream>>>(
            cur, W4 + (size_t)r * FD * 2, s_out + (size_t)r * N, h4, N);
        scatter2<<<(E + 255) / 256, 256, 0, stream>>>(
            h4, src + (size_t)r * E, dst + (size_t)r * E,
            s_in + (size_t)r * N, outp, E);
    }
    bias2<<<(2 * N + 255) / 256, 256, 0, stream>>>(outp, b4, 2 * N);
}